// GAT_63556926046387
// MI455X (gfx1250) — compile-verified
//
#include <hip/hip_runtime.h>

#define NN 50000
#define NE 800000
#define FD 64
#define ET (NE + NN)   // edges + self loops

typedef __attribute__((ext_vector_type(2))) float v2f;
typedef __attribute__((ext_vector_type(8))) float v8f;

// Monotone float -> uint mapping so unsigned atomicMax == float max (handles negatives / -inf)
__device__ __forceinline__ unsigned f2ord(float f) {
    unsigned b = __float_as_uint(f);
    return (b & 0x80000000u) ? ~b : (b | 0x80000000u);
}
__device__ __forceinline__ float ord2f(unsigned u) {
    return (u & 0x80000000u) ? __uint_as_float(u & 0x7FFFFFFFu) : __uint_as_float(~u);
}

__device__ __forceinline__ float lrelu(float v) { return v > 0.f ? v : 0.2f * v; }

// ---------------- zero init: acc[N*D], s[N], m[N] ----------------
__global__ void k_zero(float* __restrict__ acc, float* __restrict__ s, unsigned* __restrict__ m) {
    int i = blockIdx.x * blockDim.x + threadIdx.x;
    if (i < NN * FD) acc[i] = 0.f;
    if (i < NN) { s[i] = 0.f; m[i] = 0u; }
}

// ---------------- H = X @ W via V_WMMA_F32_16X16X4_F32 ----------------
// One wave per 16x16 output tile. Block = 128 threads = 4 waves = 4 column tiles (FD=64).
// grid.x = NN/16 row tiles (exact).
__global__ __launch_bounds__(128) void k_gemm_wmma(const float* __restrict__ X,
                                                   const float* __restrict__ W,
                                                   float* __restrict__ H) {
    const int lane = threadIdx.x & 31;
    const int wave = threadIdx.x >> 5;
    const int half = lane >> 4;     // 0: lanes 0-15, 1: lanes 16-31
    const int l16  = lane & 15;
    const int rowBase = blockIdx.x * 16;
    const int colBase = wave * 16;

    v8f c = {};
    const float* xrow = X + (size_t)(rowBase + l16) * FD;  // A: row m = l16
    #pragma unroll
    for (int k = 0; k < FD / 4; ++k) {
        const int kb = k * 4 + 2 * half;  // ISA 16x4 f32 A layout: half selects K pair
        v2f a, b;
        a.x = xrow[kb];
        a.y = xrow[kb + 1];
        b.x = W[kb * FD + colBase + l16];         // B: 4x16, row-striped across lanes
        b.y = W[(kb + 1) * FD + colBase + l16];
        c = __builtin_amdgcn_wmma_f32_16x16x4_f32(false, a, false, b, (short)0, c,
                                                  false, false);
    }
    // C/D layout: VGPR i -> M = i + 8*half, N = l16
    float* hout = H + colBase + l16;
    #pragma unroll
    for (int i = 0; i < 8; ++i)
        hout[(size_t)(rowBase + i + 8 * half) * FD] = c[i];
}

// ---------------- per-node attention dots: alpha_s = h.a_src, alpha_d = h.a_dst ----------------
__global__ void k_dots(const float* __restrict__ H, const float* __restrict__ av_s,
                       const float* __restrict__ av_d, float* __restrict__ alps,
                       float* __restrict__ alpd) {
    int n = blockIdx.x * blockDim.x + threadIdx.x;
    if (n >= NN) return;
    const float4* hp = (const float4*)(H + (size_t)n * FD);
    const float4* p4 = (const float4*)av_s;
    const float4* q4 = (const float4*)av_d;
    float s1 = 0.f, s2 = 0.f;
    #pragma unroll
    for (int i = 0; i < FD / 4; ++i) {
        float4 v = hp[i], p = p4[i], q = q4[i];
        s1 += v.x * p.x + v.y * p.y + v.z * p.z + v.w * p.w;
        s2 += v.x * q.x + v.y * q.y + v.z * q.z + v.w * q.w;
    }
    alps[n] = s1;
    alpd[n] = s2;
}

__device__ __forceinline__ void edge_sd(const int* __restrict__ ei, int e, int& src, int& dst) {
    if (e < NE) { src = ei[e]; dst = ei[NE + e]; }
    else        { src = dst = e - NE; }            // implicit self loops
}

// ---------------- segment max over dst ----------------
__global__ void k_edge_max(const int* __restrict__ ei, const float* __restrict__ alps,
                           const float* __restrict__ alpd, unsigned* __restrict__ m) {
    int e = blockIdx.x * blockDim.x + threadIdx.x;
    if (e >= ET) return;
    int src, dst; edge_sd(ei, e, src, dst);
    float v = lrelu(alps[src] + alpd[dst]);
    atomicMax(&m[dst], f2ord(v));
}

// ---------------- softmax denominator ----------------
__global__ void k_edge_sum(const int* __restrict__ ei, const float* __restrict__ alps,
                           const float* __restrict__ alpd, const unsigned* __restrict__ m,
                           float* __restrict__ s) {
    int e = blockIdx.x * blockDim.x + threadIdx.x;
    if (e >= ET) return;
    int src, dst; edge_sd(ei, e, src, dst);
    float v = lrelu(alps[src] + alpd[dst]);
    atomicAdd(&s[dst], __expf(v - ord2f(m[dst])));
}

// ---------------- weighted scatter: acc[dst] += ex * h[src]; one wave per edge ----------------
__global__ __launch_bounds__(256) void k_edge_scatter(const int* __restrict__ ei,
                                                      const float* __restrict__ alps,
                                                      const float* __restrict__ alpd,
                                                      const unsigned* __restrict__ m,
                                                      const float* __restrict__ H,
                                                      float* __restrict__ acc) {
    int e = (int)(blockIdx.x * (blockDim.x >> 5) + (threadIdx.x >> 5));
    if (e >= ET) return;  // wave-uniform exit
    int lane = threadIdx.x & 31;
    int src, dst; edge_sd(ei, e, src, dst);
    float v = lrelu(alps[src] + alpd[dst]);
    float ex = __expf(v - ord2f(m[dst]));
    float2 hv = ((const float2*)(H + (size_t)src * FD))[lane];  // coalesced 256B per edge
    atomicAdd(&acc[(size_t)dst * FD + lane * 2],     ex * hv.x);
    atomicAdd(&acc[(size_t)dst * FD + lane * 2 + 1], ex * hv.y);
}

// ---------------- out = acc/(s+eps) + b, optional relu ----------------
__global__ void k_norm(const float* __restrict__ acc, const float* __restrict__ s,
                       const float* __restrict__ b, float* __restrict__ out, int relu) {
    int i = blockIdx.x * blockDim.x + threadIdx.x;
    if (i >= NN * FD) return;
    int n = i >> 6, k = i & 63;
    float v = acc[i] / (s[n] + 1e-16f) + b[k];
    if (relu) v = fmaxf(v, 0.f);
    out[i] = v;
}

static void run_layer(hipStream_t st, const float* in, const int* ei, const float* W,
                      const float* as_, const float* ad_, const float* b, float* H,
                      float* acc, float* alps, float* alpd, unsigned* m, float* s,
                      float* out, int relu) {
    k_zero<<<(NN * FD + 255) / 256, 256, 0, st>>>(acc, s, m);
    k_gemm_wmma<<<NN / 16, 128, 0, st>>>(in, W, H);
    k_dots<<<(NN + 255) / 256, 256, 0, st>>>(H, as_, ad_, alps, alpd);
    k_edge_max<<<(ET + 255) / 256, 256, 0, st>>>(ei, alps, alpd, m);
    k_edge_sum<<<(ET + 255) / 256, 256, 0, st>>>(ei, alps, alpd, m, s);
    k_edge_scatter<<<(ET + 7) / 8, 256, 0, st>>>(ei, alps, alpd, m, H, acc);
    k_norm<<<(NN * FD + 255) / 256, 256, 0, st>>>(acc, s, b, out, relu);
}

extern "C" void kernel_launch(void* const* d_in, const int* in_sizes, int n_in,
                              void* d_out, int out_size, void* d_ws, size_t ws_size,
                              hipStream_t stream) {
    (void)in_sizes; (void)n_in; (void)out_size; (void)ws_size;
    const float* x  = (const float*)d_in[0];
    const int*   ei = (const int*)d_in[1];
    const float *W1 = (const float*)d_in[2],  *a1s = (const float*)d_in[3],
                *a1d = (const float*)d_in[4], *b1  = (const float*)d_in[5];
    const float *W2 = (const float*)d_in[6],  *a2s = (const float*)d_in[7],
                *a2d = (const float*)d_in[8], *b2  = (const float*)d_in[9];
    const float *W3 = (const float*)d_in[10], *a3s = (const float*)d_in[11],
                *a3d = (const float*)d_in[12],*b3  = (const float*)d_in[13];

    float* ws   = (float*)d_ws;
    float* B0   = ws;                          // H (transformed features)
    float* B1   = B0 + (size_t)NN * FD;        // acc (unnormalized aggregation)
    float* B2   = B1 + (size_t)NN * FD;        // layer output / next input
    float* alps = B2 + (size_t)NN * FD;
    float* alpd = alps + NN;
    unsigned* m = (unsigned*)(alpd + NN);
    float* s    = (float*)(m + NN);
    float* out  = (float*)d_out;

    run_layer(stream, x,  ei, W1, a1s, a1d, b1, B0, B1,  alps, alpd, m, s, B2,  1);
    run_layer(stream, B2, ei, W2, a2s, a2d, b2, B0, B1,  alps, alpd, m, s, B2,  1);
    run_layer(stream, B2, ei, W3, a3s, a3d, b3, B0, out, alps, alpd, m, s, out, 0);
}